// MultiBoxLoss_2808908611890
// MI455X (gfx1250) — compile-verified
//
#include <hip/hip_runtime.h>
#include <hip/hip_bf16.h>
#include <math.h>

#define BDIM 256
constexpr int B = 64;
constexpr int P = 24564;
constexpr int NC = 21;      // classes
constexpr int NT = 16;      // truths per image
constexpr float THRESH = 0.5f;
constexpr float V0 = 0.1f, V1 = 0.2f;

typedef __attribute__((ext_vector_type(2))) float v2f;
typedef __attribute__((ext_vector_type(8))) float v8f;
typedef __attribute__((ext_vector_type(4))) unsigned int u32x4;
typedef __attribute__((ext_vector_type(8))) int i32x8;
typedef __attribute__((ext_vector_type(4))) int i32x4;

// ---------------- helpers ----------------

__device__ __forceinline__ unsigned long long shfl_down_u64(unsigned long long v, int off) {
  unsigned lo = (unsigned)(v & 0xFFFFFFFFull);
  unsigned hi = (unsigned)(v >> 32);
  lo = __shfl_down(lo, off, 32);
  hi = __shfl_down(hi, off, 32);
  return ((unsigned long long)hi << 32) | (unsigned long long)lo;
}

// monotonic float -> uint key (total order, handles tiny negatives)
__device__ __forceinline__ unsigned mb_okey(float f) {
  unsigned u = __float_as_uint(f);
  return (u & 0x80000000u) ? ~u : (u | 0x80000000u);
}
__device__ __forceinline__ float mb_ikey(unsigned k) {
  unsigned u = (k & 0x80000000u) ? (k & 0x7FFFFFFFu) : ~k;
  return __uint_as_float(u);
}

// ---------------- kernels ----------------

// scal layout: [0]=loss_l sum, [1]=pos CE sum, [2]=neg CE sum, [3]=num_pos total
__global__ void mb_init(unsigned long long* bpt, float* numpos_b, float* scal) {
  int i = threadIdx.x;
  if (i < B * NT) bpt[i] = 0ull;
  if (i < B) numpos_b[i] = 0.f;
  if (i < 4) scal[i] = 0.f;
}

__global__ void mb_match(const float* __restrict__ priors, const float* __restrict__ truths,
                         float* __restrict__ best_ov, int* __restrict__ best_idx,
                         unsigned long long* __restrict__ bpt) {
  __shared__ float tr[NT * 4];
  int b = blockIdx.y;
  int tid = threadIdx.x;
  if (tid < NT * 4) tr[tid] = truths[b * NT * 4 + tid];
  __syncthreads();

  int p = blockIdx.x * BDIM + tid;
  bool ok = p < P;
  int pc = ok ? p : P - 1;
  const float4 pr = *reinterpret_cast<const float4*>(priors + (size_t)pc * 4);
  float px1 = pr.x - 0.5f * pr.z, py1 = pr.y - 0.5f * pr.w;
  float px2 = pr.x + 0.5f * pr.z, py2 = pr.y + 0.5f * pr.w;
  float ap = pr.z * pr.w;

  float bov = -1.f;
  int bidx = 0;
#pragma unroll
  for (int n = 0; n < NT; ++n) {
    float tx1 = tr[n * 4 + 0], ty1 = tr[n * 4 + 1];
    float tx2 = tr[n * 4 + 2], ty2 = tr[n * 4 + 3];
    float ix = fmaxf(fminf(px2, tx2) - fmaxf(px1, tx1), 0.f);
    float iy = fmaxf(fminf(py2, ty2) - fmaxf(py1, ty1), 0.f);
    float inter = ix * iy;
    float at = (tx2 - tx1) * (ty2 - ty1);
    float iou = inter / (ap + at - inter);
    if (iou > bov) { bov = iou; bidx = n; }   // first max wins (argmax over N)

    // per-truth argmax over priors: iou bits high, (P-1-p) low so ties -> smallest p
    unsigned key = __float_as_uint(iou);       // iou >= 0 -> monotonic bits
    unsigned long long pack =
        ok ? (((unsigned long long)key << 32) | (unsigned)(P - 1 - p)) : 0ull;
    for (int off = 16; off > 0; off >>= 1) {
      unsigned long long o = shfl_down_u64(pack, off);
      pack = (o > pack) ? o : pack;
    }
    if ((tid & 31) == 0) atomicMax(&bpt[b * NT + n], pack);
  }
  if (ok) {
    best_ov[(size_t)b * P + p] = bov;
    best_idx[(size_t)b * P + p] = bidx;
  }
}

__global__ void mb_override(const unsigned long long* __restrict__ bpt,
                            float* __restrict__ best_ov, int* __restrict__ best_idx) {
  int i = blockIdx.x * blockDim.x + threadIdx.x;
  if (i >= B * NT) return;
  int b = i / NT, n = i % NT;
  unsigned long long v = bpt[i];
  int p = P - 1 - (int)(v & 0xFFFFFFFFull);
  best_ov[(size_t)b * P + p] = 2.0f;
  best_idx[(size_t)b * P + p] = n;
}

__global__ void mb_loss(const float* __restrict__ loc_data, const float* __restrict__ conf_data,
                        const float* __restrict__ priors, const float* __restrict__ truths,
                        const int* __restrict__ labels,
                        const float* __restrict__ best_ov, const int* __restrict__ best_idx,
                        float* __restrict__ ce_mine, float* __restrict__ numpos_b,
                        float* __restrict__ scal) {
  extern __shared__ char smem[];
  float* ctile = (float*)smem;                               // BDIM*NC floats at LDS 0
  float* red = (float*)(smem + BDIM * NC * sizeof(float));   // 3*BDIM floats

  int b = blockIdx.y, tid = threadIdx.x;
  int p0 = blockIdx.x * BDIM;

  // --- TDM: DMA the 256x21 f32 conf tile into LDS (2D descriptor) ---
  if (tid == 0) {
    unsigned long long ga =
        (unsigned long long)(uintptr_t)(conf_data + ((size_t)b * P + p0) * NC);
    unsigned rows = (unsigned)(P - p0);   // remaining rows: OOB tile rows read as zero
    u32x4 g0;
    g0[0] = 1u;                                            // count=1, user descriptor
    g0[1] = 0u;                                            // lds_addr = dynamic-LDS base
    g0[2] = (unsigned)(ga & 0xFFFFFFFFull);                // global_addr[31:0]
    g0[3] = (unsigned)((ga >> 32) & 0x01FFFFFFull) | (2u << 30);  // addr[56:32], type=2
    i32x8 g1;
    g1[0] = (int)(2u << 16);                               // data_size = 4B
    g1[1] = (int)((unsigned)NC << 16);                     // tensor_dim0[15:0]=21
    g1[2] = (int)((rows & 0xFFFFu) << 16);                 // dim0 hi=0 | tensor_dim1 lo
    g1[3] = (int)(((rows >> 16) & 0xFFFFu) | ((unsigned)NC << 16)); // dim1 hi | tile_dim0=21
    g1[4] = (int)BDIM;                                     // tile_dim1=256 (tile_dim2=0)
    g1[5] = (int)NC;                                       // tensor_dim0_stride=21
    g1[6] = 0;
    g1[7] = 0;
    i32x4 gz = {0, 0, 0, 0};
#if __has_include(<hip/amd_detail/amd_gfx1250_TDM.h>)
    i32x8 gz8 = {0, 0, 0, 0, 0, 0, 0, 0};
    __builtin_amdgcn_tensor_load_to_lds(g0, g1, gz, gz, gz8, 0);
#else
    __builtin_amdgcn_tensor_load_to_lds(g0, g1, gz, gz, 0);
#endif
    __builtin_amdgcn_s_wait_tensorcnt(0);
  }
  __syncthreads();

  int p = p0 + tid;
  bool ok = p < P;
  int pc = ok ? p : P - 1;

  // logsumexp over 21 classes from LDS (stride 21 -> conflict-free banks)
  float m = -1e30f;
#pragma unroll
  for (int c = 0; c < NC; ++c) m = fmaxf(m, ctile[tid * NC + c]);
  float se = 0.f;
#pragma unroll
  for (int c = 0; c < NC; ++c) se += __expf(ctile[tid * NC + c] - m);
  float lse = m + __logf(se);

  float ov = best_ov[(size_t)b * P + pc];
  int ti = best_idx[(size_t)b * P + pc];
  int lab = labels[b * NT + ti];
  int conf_t = (ov < THRESH) ? 0 : (lab + 1);
  bool pos = ok && (conf_t > 0);
  float ce = lse - ctile[tid * NC + conf_t];
  if (ok) ce_mine[(size_t)b * P + p] = pos ? 0.f : ce;
  float cep = pos ? ce : 0.f;

  // localization target encode + smooth-L1 (vector loads: all rows 16B aligned)
  const float4 pr = *reinterpret_cast<const float4*>(priors + (size_t)pc * 4);
  const float4 tb = *reinterpret_cast<const float4*>(truths + (size_t)(b * NT + ti) * 4);
  const float4 ld = *reinterpret_cast<const float4*>(loc_data + ((size_t)b * P + pc) * 4);
  float tgt0 = (0.5f * (tb.x + tb.z) - pr.x) / (V0 * pr.z);
  float tgt1 = (0.5f * (tb.y + tb.w) - pr.y) / (V0 * pr.w);
  float tgt2 = __logf((tb.z - tb.x) / pr.z) / V1;
  float tgt3 = __logf((tb.w - tb.y) / pr.w) / V1;
  float df[4] = {ld.x - tgt0, ld.y - tgt1, ld.z - tgt2, ld.w - tgt3};
  float sl1 = 0.f;
#pragma unroll
  for (int c2 = 0; c2 < 4; ++c2) {
    float ad = fabsf(df[c2]);
    sl1 += (ad < 1.f) ? (0.5f * df[c2] * df[c2]) : (ad - 0.5f);
  }
  sl1 = pos ? sl1 : 0.f;

  // --- fused triple block-sum through the matrix pipe (one barrier) ---
  red[tid] = sl1;
  red[BDIM + tid] = cep;
  red[2 * BDIM + tid] = pos ? 1.f : 0.f;
  __syncthreads();
  if (tid < 32) {                 // wave 0, EXEC all ones (WMMA requirement)
    v2f ones; ones[0] = 1.f; ones[1] = 1.f;
    float r[3];
#pragma unroll
    for (int s2 = 0; s2 < 3; ++s2) {
      v8f acc = {0.f, 0.f, 0.f, 0.f, 0.f, 0.f, 0.f, 0.f};
#pragma unroll
      for (int c = 0; c < BDIM / 64; ++c) {
        v2f a;
        a[0] = red[s2 * BDIM + c * 64 + tid];
        a[1] = red[s2 * BDIM + c * 64 + 32 + tid];
        // A = 64 partials (16x4), B = ones(4x16): sum(D) == 16*sum(A), layout-agnostic
        acc = __builtin_amdgcn_wmma_f32_16x16x4_f32(false, a, false, ones,
                                                    (short)0, acc, false, false);
      }
      float s = 0.f;
#pragma unroll
      for (int rr = 0; rr < 8; ++rr) s += acc[rr];
#pragma unroll
      for (int off = 16; off > 0; off >>= 1) s += __shfl_down(s, off, 32);
      r[s2] = s * (1.0f / 16.0f);
    }
    if (tid == 0) {
      atomicAdd(&scal[0], r[0]);
      atomicAdd(&scal[1], r[1]);
      atomicAdd(&numpos_b[b], r[2]);
      atomicAdd(&scal[3], r[2]);
    }
  }
}

// one block per batch: MSB-first 256-bin radix select of the k-th largest key,
// then sum of the top-k negative CE values. 5 sweeps of the row total.
__global__ void mb_mine(const float* __restrict__ ce_mine, const float* __restrict__ numpos_b,
                        float* __restrict__ scal) {
  __shared__ unsigned hist[256];
  __shared__ unsigned s_prefix;
  __shared__ int s_k;
  __shared__ float s2[BDIM / 32];
  __shared__ int c2[BDIM / 32];
  int b = blockIdx.x, tid = threadIdx.x;
  int np = (int)(numpos_b[b] + 0.5f);
  int k = min(3 * np, P - 1);
  if (k <= 0) return;  // uniform across block

  const float* row = ce_mine + (size_t)b * P;
  if (tid == 0) { s_prefix = 0u; s_k = k; }

#pragma unroll
  for (int d = 3; d >= 0; --d) {
    __syncthreads();                 // publish s_prefix/s_k, retire old hist readers
    hist[tid] = 0u;                  // BDIM == 256
    __syncthreads();
    unsigned pre = s_prefix;
    for (int i = tid; i < P; i += BDIM) {
      unsigned key = mb_okey(row[i]);
      bool in_bucket = (d == 3) || ((key >> (d * 8 + 8)) == pre);
      if (in_bucket) atomicAdd(&hist[(key >> (d * 8)) & 0xFFu], 1u);
    }
    __syncthreads();
    if (tid == 0) {
      int kk = s_k;
      unsigned acc = 0;
      for (int j = 255; j >= 0; --j) {
        unsigned h = hist[j];
        if ((int)(acc + h) >= kk) {       // bucket containing the kk-th largest
          s_k = kk - (int)acc;
          s_prefix = (pre << 8) | (unsigned)j;
          break;
        }
        acc += h;
      }
    }
  }
  __syncthreads();
  unsigned T = s_prefix;               // key of the k-th largest value

  // sum strictly-greater values, pad ties at the k-th value
  int cgt = 0;
  float sgt = 0.f;
  for (int i = tid; i < P; i += BDIM) {
    float v = row[i];
    if (mb_okey(v) > T) { cgt++; sgt += v; }
  }
  for (int off = 16; off > 0; off >>= 1) {
    cgt += __shfl_down(cgt, off, 32);
    sgt += __shfl_down(sgt, off, 32);
  }
  if ((tid & 31) == 0) { s2[tid >> 5] = sgt; c2[tid >> 5] = cgt; }
  __syncthreads();
  if (tid == 0) {
    float S = 0.f;
    int Cg = 0;
    for (int w = 0; w < BDIM / 32; ++w) { S += s2[w]; Cg += c2[w]; }
    S += (float)(k - Cg) * mb_ikey(T);
    atomicAdd(&scal[2], S);
  }
}

__global__ void mb_fin(const float* __restrict__ scal, float* __restrict__ out) {
  float n = scal[3];
  out[0] = scal[0] / n;
  out[1] = (scal[1] + scal[2]) / n;
}

// ---------------- launch ----------------

extern "C" void kernel_launch(void* const* d_in, const int* in_sizes, int n_in,
                              void* d_out, int out_size, void* d_ws, size_t ws_size,
                              hipStream_t stream) {
  const float* loc_data = (const float*)d_in[0];   // [B,P,4]
  const float* conf_data = (const float*)d_in[1];  // [B,P,C]
  const float* priors = (const float*)d_in[2];     // [P,4]
  const float* truths = (const float*)d_in[3];     // [B,N,4]
  const int* labels = (const int*)d_in[4];         // [B,N]

  char* ws = (char*)d_ws;
  size_t off = 0;
  auto alloc = [&](size_t bytes) {
    char* p = ws + off;
    off = (off + bytes + 255) & ~(size_t)255;
    return p;
  };
  float* best_ov = (float*)alloc((size_t)B * P * sizeof(float));
  int* best_idx = (int*)alloc((size_t)B * P * sizeof(int));
  float* ce_mine = (float*)alloc((size_t)B * P * sizeof(float));
  unsigned long long* bpt = (unsigned long long*)alloc((size_t)B * NT * 8);
  float* numpos_b = (float*)alloc((size_t)B * sizeof(float));
  float* scal = (float*)alloc(4 * sizeof(float));
  (void)ws_size; (void)in_sizes; (void)n_in; (void)out_size;

  mb_init<<<1, 1024, 0, stream>>>(bpt, numpos_b, scal);

  dim3 grid((P + BDIM - 1) / BDIM, B);
  mb_match<<<grid, BDIM, 0, stream>>>(priors, truths, best_ov, best_idx, bpt);
  mb_override<<<(B * NT + BDIM - 1) / BDIM, BDIM, 0, stream>>>(bpt, best_ov, best_idx);

  size_t shm = (size_t)BDIM * NC * sizeof(float) + (size_t)3 * BDIM * sizeof(float);
  mb_loss<<<grid, BDIM, shm, stream>>>(loc_data, conf_data, priors, truths, labels,
                                       best_ov, best_idx, ce_mine, numpos_b, scal);
  mb_mine<<<B, BDIM, 0, stream>>>(ce_mine, numpos_b, scal);
  mb_fin<<<1, 1, 0, stream>>>(scal, (float*)d_out);
}